// EdgeConvGAT_79216376808178
// MI455X (gfx1250) — compile-verified
//
#include <hip/hip_runtime.h>
#include <hip/hip_bf16.h>

// ---------------------------------------------------------------------------
// EdgeConv-GAT fused kernel for MI455X (gfx1250, wave32).
//
// Roofline: ~63 GFLOP fp32 projections + ~5 GFLOP attention; unique traffic
// ~10 MB (h = 5 MB, W = 384 KB) -> L2 resident. Compute-bound fp32 -> keep
// ALL matmuls (projections, q@k^T, alpha@v) on V_WMMA_F32_16X16X4_F32
// (exact fp32 mul/add, matches the fp32 reference up to reduction order).
// Neighbor gather staged into LDS with GLOBAL_LOAD_ASYNC_TO_LDS_B128
// (ASYNCcnt-tracked DMA, CDNA5 async path).
// ---------------------------------------------------------------------------

typedef __attribute__((ext_vector_type(2))) float v2f;
typedef __attribute__((ext_vector_type(8))) float v8f;

__device__ __forceinline__ v8f wmma_f32_4(v2f a, v2f b, v8f c) {
  // 8 args: (neg_a, A, neg_b, B, c_mod, C, reuse_a, reuse_b)
  return __builtin_amdgcn_wmma_f32_16x16x4_f32(
      false, a, false, b, (short)0, c, false, false);
}

namespace cfg {
constexpr int D      = 32;    // neighbors per node
constexpr int F      = 128;   // feature dim
constexpr int C      = 128;   // channels
constexpr int H      = 4;     // heads
constexpr int HD     = 32;    // head dim
constexpr int TWOF   = 256;   // 2*F (K dim of projections)
constexpr int KSTEPS = TWOF / 4;   // 64 WMMA K-steps (projections)
constexpr int NTILES = C / 16;     // 8
constexpr int PAIRS  = 3 * NTILES; // 24 (Wq/Wk/Wv x ntile)
// LDS strides padded to spread row-strided WMMA fragment reads over 64 banks.
constexpr int FEAT_STRIDE = TWOF + 4; // 260 (260 % 64 == 4)
constexpr int QKV_STRIDE  = C + 4;    // 132
constexpr int CTX_STRIDE  = C + 4;    // 132 (aliased onto feat buffer)
constexpr int SC_STRIDE   = D + 4;    // 36  (36 % 64 == 36, gcd(36,64)=4)
}  // namespace cfg

// ---------------------------------------------------------------------------
// Pack Wq/Wk/Wv (row-major 256x128) into B-fragment order so the WMMA inner
// loop does ONE coalesced b64 load per fragment:
//   pack[((pp*KSTEPS+ks)*32+lane)*2 + j] = W[ks*4+2*(lane>>4)+j][ntile*16+(lane&15)]
// with pp = wmat*8 + ntile.
// ---------------------------------------------------------------------------
__global__ void EdgeGAT_pack_w(const float* __restrict__ Wq,
                               const float* __restrict__ Wk,
                               const float* __restrict__ Wv,
                               float* __restrict__ pack) {
  using namespace cfg;
  int gid = blockIdx.x * 256 + threadIdx.x;
  if (gid >= PAIRS * KSTEPS * 32) return;
  int lane  = gid & 31;
  int ks    = (gid >> 5) & (KSTEPS - 1);
  int pp    = gid >> 11;
  int wmat  = pp >> 3;
  int ntile = pp & 7;
  const float* W = (wmat == 0) ? Wq : (wmat == 1) ? Wk : Wv;
  int col = ntile * 16 + (lane & 15);
  int kb  = ks * 4 + 2 * (lane >> 4);
  pack[gid * 2 + 0] = W[kb * C + col];
  pack[gid * 2 + 1] = W[(kb + 1) * C + col];
}

// ---------------------------------------------------------------------------
// Main fused kernel: one node per workgroup (256 threads = 8 waves).
// ---------------------------------------------------------------------------
template <bool PACKED>
__global__ __launch_bounds__(256) void EdgeGAT_node_kernel(
    const float* __restrict__ h, const int* __restrict__ src_idx,
    const float* __restrict__ drop_mask,
    const float* __restrict__ Wq, const float* __restrict__ bq,
    const float* __restrict__ Wk, const float* __restrict__ bk,
    const float* __restrict__ Wv, const float* __restrict__ bv,
    const float* __restrict__ pack, float* __restrict__ out, int n_nodes) {
  using namespace cfg;

  __shared__ float stageS[D * F];              // 16,384 B raw neighbor rows
  __shared__ float featS[D * FEAT_STRIDE];     // 33,280 B (reused as ctx)
  __shared__ float qkvS[3][D * QKV_STRIDE];    // 50,688 B
  __shared__ float scS[H * D * SC_STRIDE];     // 18,432 B scores/alpha
  __shared__ float xdstS[F];
  __shared__ float maskS[D];
  __shared__ float ewS[D];
  __shared__ int   idxS[D];

  const int node = blockIdx.x;
  if (node >= n_nodes) return;
  const int tid    = threadIdx.x;
  const int lane   = tid & 31;
  const int wave   = tid >> 5;
  const int laneLo = lane & 15;
  const int laneHi = lane >> 4;

  // ------------------- Phase A0: indices, mask, ew, x_dst ------------------
  if (tid < D) {
    int s = src_idx[node * D + tid];
    idxS[tid]  = s;
    maskS[tid] = drop_mask[node * D + tid];
    float dist = (float)(node - s);
    // sign(log(dist)-log(9)) == compare vs 9 (log monotone; log(0)=-inf -> -1)
    ewS[tid] = (dist > 9.0f) ? 1.0f : ((dist == 9.0f) ? 0.0f : -1.0f);
  }
  if (tid < F) xdstS[tid] = h[node * F + tid];
  __syncthreads();

  // --------- Phase A1: async DMA gather of neighbor rows into LDS ----------
  // 32 rows x 512 B = 1024 x b128 chunks; each lane DMAs 16 B per issue.
#pragma unroll
  for (int i = 0; i < (D * F) / (256 * 4); ++i) {  // 4 issues/thread
    int chunk = tid + i * 256;                     // b128 chunk id
    int d  = chunk >> 5;                           // 32 chunks per row
    int c4 = chunk & 31;
    unsigned long long ga =
        (unsigned long long)(const void*)(h + (size_t)idxS[d] * F + c4 * 4);
    unsigned int la =
        (unsigned int)(unsigned long long)(const void*)(&stageS[d * F + c4 * 4]);
    asm volatile("global_load_async_to_lds_b128 %0, %1, off"
                 :: "v"(la), "v"(ga)
                 : "memory");
  }
  asm volatile("s_wait_asynccnt 0x0" ::: "memory");
  __syncthreads();

  // ----------------------- Phase A2: build feat in LDS ---------------------
#pragma unroll
  for (int i = 0; i < (D * F) / 256; ++i) {  // 16 passes
    int e = tid + i * 256;
    int d = e >> 7, c = e & (F - 1);
    float xs = stageS[d * F + c];
    float xd = xdstS[c];
    featS[d * FEAT_STRIDE + c]     = xs - xd;  // x_src - x_dst
    featS[d * FEAT_STRIDE + F + c] = xd;       // x_dst
  }
  __syncthreads();

  // -------------------- Phase B: q/k/v projections (WMMA) ------------------
  for (int pi = 0; pi < 3; ++pi) {
    int pp    = wave * 3 + pi;
    int wmat  = pp >> 3;
    int ntile = pp & 7;
    const float* Wsel = (wmat == 0) ? Wq : (wmat == 1) ? Wk : Wv;
    const float* bsel = (wmat == 0) ? bq : (wmat == 1) ? bk : bv;
    int col = ntile * 16 + laneLo;

    v8f acc0 = {};  // rows 0..15
    v8f acc1 = {};  // rows 16..31
#pragma unroll 4
    for (int ks = 0; ks < KSTEPS; ++ks) {
      int kb = ks * 4 + 2 * laneHi;
      v2f bfrag;
      if (PACKED) {
        const float* p = pack + (((size_t)pp * KSTEPS + ks) * 32 + lane) * 2;
        bfrag = *(const v2f*)p;  // one coalesced b64 per lane
      } else {
        bfrag.x = Wsel[kb * C + col];
        bfrag.y = Wsel[(kb + 1) * C + col];
      }
      v2f a0 = *(const v2f*)&featS[(0 * 16 + laneLo) * FEAT_STRIDE + kb];
      v2f a1 = *(const v2f*)&featS[(1 * 16 + laneLo) * FEAT_STRIDE + kb];
      acc0 = wmma_f32_4(a0, bfrag, acc0);
      acc1 = wmma_f32_4(a1, bfrag, acc1);
    }
    float bias = bsel[col];
#pragma unroll
    for (int r = 0; r < 8; ++r) {
      int d0  = r + 8 * laneHi;
      qkvS[wmat][(d0)      * QKV_STRIDE + col] = (acc0[r] + bias) * maskS[d0];
      qkvS[wmat][(16 + d0) * QKV_STRIDE + col] = (acc1[r] + bias) * maskS[16 + d0];
    }
  }
  __syncthreads();

  // ---------------- Phase C1: attention scores q@k^T (WMMA) ----------------
  // 16 tiles: t = head*4 + mtile*2 + ntile; wave w owns t = 2w, 2w+1
  // (same head & mtile -> shared A fragment, ntile = 0/1).
  {
    const float scale = 0.08838834764831845f;  // 1/sqrt(128)
    int t0    = 2 * wave;
    int head  = t0 >> 2;
    int mtile = (t0 >> 1) & 1;
    v8f acc0 = {};
    v8f acc1 = {};
#pragma unroll
    for (int ks = 0; ks < HD / 4; ++ks) {  // 8 steps over head dim
      int kb = ks * 4 + 2 * laneHi;
      v2f a  = *(const v2f*)&qkvS[0][(mtile * 16 + laneLo) * QKV_STRIDE +
                                     head * HD + kb];
      v2f b0 = *(const v2f*)&qkvS[1][(0 * 16 + laneLo) * QKV_STRIDE +
                                     head * HD + kb];
      v2f b1 = *(const v2f*)&qkvS[1][(1 * 16 + laneLo) * QKV_STRIDE +
                                     head * HD + kb];
      acc0 = wmma_f32_4(a, b0, acc0);
      acc1 = wmma_f32_4(a, b1, acc1);
    }
#pragma unroll
    for (int r = 0; r < 8; ++r) {
      int m = mtile * 16 + r + 8 * laneHi;
      scS[(head * D + m) * SC_STRIDE + (0 * 16 + laneLo)] = acc0[r] * scale;
      scS[(head * D + m) * SC_STRIDE + (1 * 16 + laneLo)] = acc1[r] * scale;
    }
  }
  __syncthreads();

  // ------------- Phase C2: softmax + post-softmax ew (VALU, tiny) ----------
  if (tid < H * D) {  // row = head*32 + qi == tid
    float s[D];
#pragma unroll
    for (int ki = 0; ki < D; ++ki) s[ki] = scS[tid * SC_STRIDE + ki];
    float m = s[0];
#pragma unroll
    for (int ki = 1; ki < D; ++ki) m = fmaxf(m, s[ki]);
    float sum = 0.f;
#pragma unroll
    for (int ki = 0; ki < D; ++ki) {
      s[ki] = __expf(s[ki] - m);
      sum += s[ki];
    }
    float inv = 1.0f / sum;
#pragma unroll
    for (int ki = 0; ki < D; ++ki)
      scS[tid * SC_STRIDE + ki] = s[ki] * inv * ewS[ki];  // alpha * ew
  }
  __syncthreads();

  // ------------------- Phase C3: ctx = alpha @ v (WMMA) --------------------
  float* ctx = featS;  // feat is dead; alias as ctx[d][c] with CTX_STRIDE
  {
    int t0    = 2 * wave;
    int head  = t0 >> 2;
    int mtile = (t0 >> 1) & 1;
    v8f acc0 = {};
    v8f acc1 = {};
#pragma unroll
    for (int ks = 0; ks < D / 4; ++ks) {  // 8 steps over key dim
      int kb = ks * 4 + 2 * laneHi;
      v2f a = *(const v2f*)&scS[(head * D + mtile * 16 + laneLo) * SC_STRIDE + kb];
      v2f b0, b1;  // B[key][hd]: rows kb, kb+1 of v
      b0.x = qkvS[2][(kb)     * QKV_STRIDE + head * HD + 0 * 16 + laneLo];
      b0.y = qkvS[2][(kb + 1) * QKV_STRIDE + head * HD + 0 * 16 + laneLo];
      b1.x = qkvS[2][(kb)     * QKV_STRIDE + head * HD + 1 * 16 + laneLo];
      b1.y = qkvS[2][(kb + 1) * QKV_STRIDE + head * HD + 1 * 16 + laneLo];
      acc0 = wmma_f32_4(a, b0, acc0);
      acc1 = wmma_f32_4(a, b1, acc1);
    }
#pragma unroll
    for (int r = 0; r < 8; ++r) {
      int m = mtile * 16 + r + 8 * laneHi;  // = neighbor index d
      ctx[m * CTX_STRIDE + head * HD + 0 * 16 + laneLo] = acc0[r];
      ctx[m * CTX_STRIDE + head * HD + 1 * 16 + laneLo] = acc1[r];
    }
  }
  __syncthreads();

  // ------------------------ Phase D: mean over neighbors -------------------
  if (tid < C) {
    float acc = 0.f;
#pragma unroll
    for (int d = 0; d < D; ++d) acc += ctx[d * CTX_STRIDE + tid];
    out[(size_t)node * C + tid] = acc * (1.0f / D);
  }
}

// ---------------------------------------------------------------------------
extern "C" void kernel_launch(void* const* d_in, const int* in_sizes, int n_in,
                              void* d_out, int out_size, void* d_ws,
                              size_t ws_size, hipStream_t stream) {
  using namespace cfg;
  const float* h       = (const float*)d_in[0];
  const int*   src_idx = (const int*)d_in[1];
  const float* drop    = (const float*)d_in[2];
  const float* Wq      = (const float*)d_in[3];
  const float* bq      = (const float*)d_in[4];
  const float* Wk      = (const float*)d_in[5];
  const float* bk      = (const float*)d_in[6];
  const float* Wv      = (const float*)d_in[7];
  const float* bv      = (const float*)d_in[8];
  float*       out     = (float*)d_out;

  const int n_nodes = in_sizes[0] / F;  // h is (N, F)

  const size_t packBytes = (size_t)PAIRS * KSTEPS * 64 * sizeof(float);  // 384 KB
  const bool packed = (d_ws != nullptr) && (ws_size >= packBytes);

  if (packed) {
    int total = PAIRS * KSTEPS * 32;
    EdgeGAT_pack_w<<<(total + 255) / 256, 256, 0, stream>>>(
        Wq, Wk, Wv, (float*)d_ws);
    EdgeGAT_node_kernel<true><<<n_nodes, 256, 0, stream>>>(
        h, src_idx, drop, Wq, bq, Wk, bk, Wv, bv, (const float*)d_ws, out,
        n_nodes);
  } else {
    EdgeGAT_node_kernel<false><<<n_nodes, 256, 0, stream>>>(
        h, src_idx, drop, Wq, bq, Wk, bk, Wv, bv, nullptr, out, n_nodes);
  }
}